// GraphConvolutionSharingClusterWeights_52536039965274
// MI455X (gfx1250) — compile-verified
//
#include <hip/hip_runtime.h>

#define IN_DIM 256
#define OUT_DIM 64

typedef __attribute__((ext_vector_type(2))) float v2f;
typedef __attribute__((ext_vector_type(8))) float v8f;

// ---------------------------------------------------------------------------
// GEMM: h[n_nodes,64] = x[n_nodes,256] @ w[256,64] via V_WMMA_F32_16X16X4_F32
// Block = 256 threads = 8 wave32's; each wave computes a 32x64 output tile
// (two 16-row WMMA tiles sharing B fragments -> 1 ds_load_b64 per 2 WMMAs).
//
// W is staged in LDS interleaved as wl[(k>>2)*256 + n*4 + (k&3)] so that a
// lane's B pair {w[k+koff][n], w[k+koff+1][n]} is one aligned 8-byte load
// (ds_load_b64 straight into an even VGPR pair, no v_mov shuffling), and the
// koff=0 half-wave hits banks {4n,4n+1} while koff=2 hits {4n+2,4n+3}:
// conflict-free. Exactly 64 KB of the 320 KB WGP LDS.
// ---------------------------------------------------------------------------
__global__ void __launch_bounds__(256)
gcn_gemm_wmma_kernel(const float* __restrict__ x, const float* __restrict__ w,
                     float* __restrict__ h, int n_nodes) {
  __shared__ float wlds[IN_DIM * OUT_DIM];  // 64 KB, interleaved layout

  const int tid = threadIdx.x;
  for (int idx = tid; idx < IN_DIM * OUT_DIM; idx += 256) {
    const int k = idx >> 6;       // 0..255
    const int n = idx & 63;       // 0..63
    wlds[(k >> 2) * 256 + n * 4 + (k & 3)] = w[idx];
  }
  __syncthreads();

  const int wave = tid >> 5;                 // wave32
  const int lane = tid & 31;
  const int rowBase = (blockIdx.x * 8 + wave) * 32;   // 32 rows per wave
  const int m = lane & 15;
  const int koff = (lane >> 4) << 1;         // lanes 0-15: K 0,1 ; lanes 16-31: K 2,3

  int arow0 = rowBase + m;
  int arow1 = rowBase + 16 + m;
  if (arow0 >= n_nodes) arow0 = n_nodes - 1;  // clamp; stores guarded below
  if (arow1 >= n_nodes) arow1 = n_nodes - 1;
  const float* __restrict__ xrow0 = x + (size_t)arow0 * IN_DIM;
  const float* __restrict__ xrow1 = x + (size_t)arow1 * IN_DIM;

  v8f acc00 = {}, acc01 = {}, acc02 = {}, acc03 = {};  // rows rowBase..+15
  v8f acc10 = {}, acc11 = {}, acc12 = {}, acc13 = {};  // rows rowBase+16..+31

#pragma unroll 4
  for (int k = 0; k < IN_DIM; k += 4) {
    const v2f a0 = *(const v2f*)(&xrow0[k + koff]);
    const v2f a1 = *(const v2f*)(&xrow1[k + koff]);

    const int base = (k >> 2) * 256 + koff;   // interleaved LDS row for this k-step
    const v2f b0 = *(const v2f*)(&wlds[base + (m +  0) * 4]);
    const v2f b1 = *(const v2f*)(&wlds[base + (m + 16) * 4]);
    const v2f b2 = *(const v2f*)(&wlds[base + (m + 32) * 4]);
    const v2f b3 = *(const v2f*)(&wlds[base + (m + 48) * 4]);

    acc00 = __builtin_amdgcn_wmma_f32_16x16x4_f32(false, a0, false, b0, (short)0, acc00, false, false);
    acc01 = __builtin_amdgcn_wmma_f32_16x16x4_f32(false, a0, false, b1, (short)0, acc01, false, false);
    acc02 = __builtin_amdgcn_wmma_f32_16x16x4_f32(false, a0, false, b2, (short)0, acc02, false, false);
    acc03 = __builtin_amdgcn_wmma_f32_16x16x4_f32(false, a0, false, b3, (short)0, acc03, false, false);
    acc10 = __builtin_amdgcn_wmma_f32_16x16x4_f32(false, a1, false, b0, (short)0, acc10, false, false);
    acc11 = __builtin_amdgcn_wmma_f32_16x16x4_f32(false, a1, false, b1, (short)0, acc11, false, false);
    acc12 = __builtin_amdgcn_wmma_f32_16x16x4_f32(false, a1, false, b2, (short)0, acc12, false, false);
    acc13 = __builtin_amdgcn_wmma_f32_16x16x4_f32(false, a1, false, b3, (short)0, acc13, false, false);
  }

  // C/D layout: VGPR r, lanes 0-15 -> (M=r, N=lane); lanes 16-31 -> (M=r+8, N=lane-16)
  const int n = lane & 15;
  const int mOff = (lane >> 4) << 3;
#pragma unroll
  for (int r = 0; r < 8; ++r) {
    const int gm0 = rowBase + mOff + r;
    if (gm0 < n_nodes) {
      float* __restrict__ hrow = h + (size_t)gm0 * OUT_DIM;
      hrow[ 0 + n] = acc00[r];
      hrow[16 + n] = acc01[r];
      hrow[32 + n] = acc02[r];
      hrow[48 + n] = acc03[r];
    }
    const int gm1 = rowBase + 16 + mOff + r;
    if (gm1 < n_nodes) {
      float* __restrict__ hrow = h + (size_t)gm1 * OUT_DIM;
      hrow[ 0 + n] = acc10[r];
      hrow[16 + n] = acc11[r];
      hrow[32 + n] = acc12[r];
      hrow[48 + n] = acc13[r];
    }
  }
}

// ---------------------------------------------------------------------------
// SpMM: out[r,:] += vals[e] * h[cols[e],:] for sorted rows.
// 64 threads per group (1 thread per feature -> 256B coalesced gathers),
// 512 contiguous edges per group, register accumulation across equal-row
// runs (avg run ~16 edges), atomic flush on row change.
// ---------------------------------------------------------------------------
#define EDGES_PER_GROUP 512

__global__ void __launch_bounds__(256)
gcn_spmm_kernel(const float* __restrict__ h, const float* __restrict__ vals,
                const int* __restrict__ rows, const int* __restrict__ cols,
                float* __restrict__ out, int n_edges) {
  const int tid = threadIdx.x;
  const int c = tid & 63;
  const long long group = (long long)blockIdx.x * 4 + (tid >> 6);
  long long e0 = group * EDGES_PER_GROUP;
  if (e0 >= n_edges) return;
  long long e1 = e0 + EDGES_PER_GROUP;
  if (e1 > n_edges) e1 = n_edges;

  int curRow = rows[e0];
  float acc = 0.0f;
  for (long long e = e0; e < e1; ++e) {
    const int r = rows[e];
    if (r != curRow) {
      atomicAdd(out + (size_t)curRow * OUT_DIM + c, acc);
      acc = 0.0f;
      curRow = r;
    }
    if (e + 16 < e1) {
      // h (25.6 MB) is L2-resident (192 MB L2): prefetch the future gather line.
      __builtin_prefetch(h + (size_t)cols[e + 16] * OUT_DIM + c, 0, 0);
    }
    acc += vals[e] * h[(size_t)cols[e] * OUT_DIM + c];
  }
  atomicAdd(out + (size_t)curRow * OUT_DIM + c, acc);
}

// ---------------------------------------------------------------------------
// Zero-init and in-place ReLU over the output.
// ---------------------------------------------------------------------------
__global__ void __launch_bounds__(256)
gcn_zero_kernel(float* __restrict__ out, int total4) {
  const int i = blockIdx.x * 256 + threadIdx.x;
  if (i < total4) {
    float4 z = make_float4(0.f, 0.f, 0.f, 0.f);
    *(float4*)(out + (size_t)i * 4) = z;
  }
}

__global__ void __launch_bounds__(256)
gcn_relu_kernel(float* __restrict__ out, int total4) {
  const int i = blockIdx.x * 256 + threadIdx.x;
  if (i < total4) {
    float4 v = *(float4*)(out + (size_t)i * 4);
    v.x = fmaxf(v.x, 0.f);
    v.y = fmaxf(v.y, 0.f);
    v.z = fmaxf(v.z, 0.f);
    v.w = fmaxf(v.w, 0.f);
    *(float4*)(out + (size_t)i * 4) = v;
  }
}

// ---------------------------------------------------------------------------
extern "C" void kernel_launch(void* const* d_in, const int* in_sizes, int n_in,
                              void* d_out, int out_size, void* d_ws, size_t ws_size,
                              hipStream_t stream) {
  const float* x    = (const float*)d_in[0];   // [n_nodes, 256]
  const float* w    = (const float*)d_in[1];   // [256, 64]
  const float* vals = (const float*)d_in[2];   // [n_edges]
  const int*   rows = (const int*)d_in[3];     // [n_edges] sorted
  const int*   cols = (const int*)d_in[4];     // [n_edges]
  float* out = (float*)d_out;                  // [n_nodes, 64]
  float* h   = (float*)d_ws;                   // [n_nodes, 64] scratch

  const int n_nodes = in_sizes[0] / IN_DIM;
  const int n_edges = in_sizes[2];
  const int total  = n_nodes * OUT_DIM;        // multiple of 4
  const int total4 = total / 4;

  gcn_zero_kernel<<<(total4 + 255) / 256, 256, 0, stream>>>(out, total4);

  gcn_gemm_wmma_kernel<<<(n_nodes + 255) / 256, 256, 0, stream>>>(x, w, h, n_nodes);

  const int ngroups = (n_edges + EDGES_PER_GROUP - 1) / EDGES_PER_GROUP;
  gcn_spmm_kernel<<<(ngroups + 3) / 4, 256, 0, stream>>>(h, vals, rows, cols, out, n_edges);

  gcn_relu_kernel<<<(total4 + 255) / 256, 256, 0, stream>>>(out, total4);
}